// StringPositionEmbedding3D_25005299597936
// MI455X (gfx1250) — compile-verified
//
#include <hip/hip_runtime.h>
#include <math.h>

typedef float v2f __attribute__((ext_vector_type(2)));
typedef float v4f __attribute__((ext_vector_type(4)));
typedef float v8f __attribute__((ext_vector_type(8)));

// Async-copy builtin operand types: b128 payload = 16-byte int vector,
// global (AS1) source pointer, LDS (AS3) destination pointer.
typedef int gvec4 __attribute__((vector_size(16)));
typedef gvec4 __attribute__((address_space(1)))* as1_v4i_ptr;
typedef gvec4 __attribute__((address_space(3)))* as3_v4i_ptr;

#define EDIM   128
#define SLEN   4096
#define LDSPAD 132   // row stride in floats: 132 mod 64 = 4 -> conflict-free, keeps 16B alignment

// ---------------------------------------------------------------------------
// Kernel 1: Cayley transform. Gauss-Jordan solve of (I+A) P = (I-A),
// A = (S - S^T)/2. Augmented 128x256 system W = [I+A | I-A] lives in d_ws.
// On exit, P[n][k] = W[n*256 + 128 + k]  (exactly the B[k][n] = P^T[k][n]
// operand for the GEMM xt = x @ P^T). One block, thread j owns column j.
// ---------------------------------------------------------------------------
__global__ __launch_bounds__(256) void cayley_solve_kernel(
    const float* __restrict__ Sc, float* __restrict__ W) {
  const int j  = threadIdx.x;       // column 0..255
  const int jj = j & 127;
  __shared__ float piv[256];
  __shared__ float colk[128];

  for (int i = 0; i < 128; ++i) {
    float a = 0.5f * (Sc[i * 128 + jj] - Sc[jj * 128 + i]);  // A[i][jj]
    float d = (i == jj) ? 1.0f : 0.0f;
    W[i * 256 + j] = (j < 128) ? (d + a) : (d - a);
  }
  __syncthreads();

  for (int k = 0; k < 128; ++k) {
    if (j < 128) colk[j] = W[j * 256 + k];     // pre-elimination column k
    float pv = W[k * 256 + k];
    piv[j] = W[k * 256 + j] / pv;              // scaled pivot row
    __syncthreads();
    float prj = piv[j];
    for (int i = 0; i < 128; ++i) {
      float w = W[i * 256 + j];
      W[i * 256 + j] = (i == k) ? prj : (w - colk[i] * prj);
    }
    __syncthreads();
  }
}

// ---------------------------------------------------------------------------
// Kernel 2: fused  xt = x @ P^T  (WMMA f32 16x16x4)  +  rotate-half RoPE.
// Block = 256 threads = 8 waves; block b owns rows [16b, 16b+16) of the
// flattened (B*H*S, 128) matrix. Wave w computes output columns [16w, 16w+16).
// ---------------------------------------------------------------------------
__global__ __launch_bounds__(256) void gemm_rope_kernel(
    const float* __restrict__ x,
    const float* __restrict__ coord,   // (SLEN)
    const float* __restrict__ W,       // P at W[n*256 + 128 + k]
    float* __restrict__ out) {
  __shared__ float sx  [16 * LDSPAD];  // staged x slab
  __shared__ float sxt [16 * LDSPAD];  // staged xt slab (cross-wave rotate-half)
  __shared__ float sfreq[64];          // RoPE inverse frequencies

  const int  t    = threadIdx.x;
  const long slab = (long)blockIdx.x * (16L * EDIM);
  const float* gx = x + slab;

  // ---- stage 16x128 slab of x into LDS ----
  const int i0 = t * 4;                // 0..1020
  const int i1 = i0 + 1024;            // 1024..2044
  float* l0 = &sx[(i0 >> 7) * LDSPAD + (i0 & 127)];
  float* l1 = &sx[(i1 >> 7) * LDSPAD + (i1 & 127)];

#if __has_builtin(__builtin_amdgcn_global_load_async_to_lds_b128)
  // CDNA5 async copy: memory -> LDS without VGPR round trip (ASYNCcnt tracked)
  __builtin_amdgcn_global_load_async_to_lds_b128(
      (as1_v4i_ptr)(gx + i0), (as3_v4i_ptr)l0, 0, 0);
  __builtin_amdgcn_global_load_async_to_lds_b128(
      (as1_v4i_ptr)(gx + i1), (as3_v4i_ptr)l1, 0, 0);
#define ASYNC_STAGED 1
#else
  *(v4f*)l0 = *(const v4f*)(gx + i0);
  *(v4f*)l1 = *(const v4f*)(gx + i1);
#endif

  // ---- block-invariant RoPE frequency table: 64 powf per block ----
  if (t < 64) sfreq[t] = powf(10000.0f, -(float)(2 * t) * (1.0f / 128.0f));

#ifdef ASYNC_STAGED
#if __has_builtin(__builtin_amdgcn_s_wait_asynccnt)
  __builtin_amdgcn_s_wait_asynccnt(0);
#else
  asm volatile("s_wait_asynccnt 0x0" ::: "memory");
#endif
#endif
  __syncthreads();

  // ---- WMMA f32 16x16x4, K = 128 -> 32 accumulating steps ----
  const int wave = t >> 5;            // 0..7 : output N-tile
  const int lane = t & 31;
  const int mrow = lane & 15;         // A layout: both lane halves hold M=0..15
  const int khi  = (lane >> 4) << 1;  // A layout: upper half-lanes hold K = k0+2, k0+3
  const int ncol = lane & 15;         // B/C/D layout: N = lane % 16

  const float* Pbase = W + (wave * 16 + ncol) * 256 + 128;  // row n of P
  v8f c = {};  // fp32 accumulator tile

#pragma unroll
  for (int k0 = 0; k0 < 128; k0 += 4) {
    v2f a = *(const v2f*)(&sx[mrow * LDSPAD + k0 + khi]);   // A[m][k0+khi .. +1]
    v2f b = *(const v2f*)(Pbase + k0 + khi);                // B[k][n] = P[n][k]
    c = __builtin_amdgcn_wmma_f32_16x16x4_f32(
        false, a, false, b, (short)0, c, false, false);
  }

  // ---- spill accumulators: VGPR r -> M = r + 8*(lane>=16), N = lane%16 ----
  const int mbase = (lane >> 4) << 3;
  const int nglob = wave * 16 + ncol;
#pragma unroll
  for (int r = 0; r < 8; ++r) {
    sxt[(mbase + r) * LDSPAD + nglob] = c[r];
  }
  __syncthreads();

  // ---- rotate-half RoPE + store (thread t: row t/16, 8 consecutive cols) ----
  const int  row  = t >> 4;
  const int  col0 = (t & 15) << 3;
  const long gm   = (long)blockIdx.x * 16 + row;      // global row in (B*H*S)
  const int  s    = (int)(gm & (SLEN - 1));           // position index (S = 4096, pow2)
  const float pos = coord[s];

  float res[8];
#pragma unroll
  for (int qp = 0; qp < 4; ++qp) {                    // one sincos per e-pair
    int   e0 = col0 + 2 * qp;
    float fr = sfreq[e0 >> 1];
    float ang = pos * fr;
    float sn, cs;
    sincosf(ang, &sn, &cs);                           // accurate: ang up to ~4096 rad
#pragma unroll
    for (int u = 0; u < 2; ++u) {
      int   e = e0 + u;
      float v = sxt[row * LDSPAD + e];
      float p = (e < 64) ? -sxt[row * LDSPAD + e + 64]
                         :  sxt[row * LDSPAD + e - 64];
      res[2 * qp + u] = v * cs + p * sn;
    }
  }
  float* go = out + gm * EDIM + col0;
  *(v4f*)(go)     = *(v4f*)(&res[0]);
  *(v4f*)(go + 4) = *(v4f*)(&res[4]);
}

// ---------------------------------------------------------------------------
extern "C" void kernel_launch(void* const* d_in, const int* in_sizes, int n_in,
                              void* d_out, int out_size, void* d_ws, size_t ws_size,
                              hipStream_t stream) {
  const float* x     = (const float*)d_in[0];   // (4,16,4096,128) f32
  const float* coord = (const float*)d_in[1];   // (4096) f32
  const float* Sc    = (const float*)d_in[2];   // (128,128) f32
  float*       out   = (float*)d_out;
  float*       W     = (float*)d_ws;            // 128x256 augmented system = 128 KiB

  cayley_solve_kernel<<<1, 256, 0, stream>>>(Sc, W);

  const int mrows  = 4 * 16 * 4096;             // 262144
  const int blocks = mrows / 16;                // 16384
  gemm_rope_kernel<<<blocks, 256, 0, stream>>>(x, coord, W, out);
}